// BertCrfForTokenClassification_1219770712220
// MI455X (gfx1250) — compile-verified
//
#include <hip/hip_runtime.h>

typedef __attribute__((ext_vector_type(2))) float v2f;
typedef __attribute__((ext_vector_type(8))) float v8f;

#define Bv 64
#define Sv 512
#define Hv 768
#define Lv 21
#define Np 32          // padded label dim (two 16-wide N tiles)
#define BS (Bv * Sv)   // 32768

// -------------------------------------------------------------------------
// Kernel 0: build padded transposed weights Wt[n][k] (n in [0,32), zeros for
// n >= 21) and padded bias, so the GEMM inner loop has zero conditionals.
// -------------------------------------------------------------------------
__global__ __launch_bounds__(256) void k_padw(const float* __restrict__ W,
                                              const float* __restrict__ bias,
                                              float* __restrict__ Wt,
                                              float* __restrict__ biasp)
{
    const int idx = blockIdx.x * 256 + threadIdx.x;
    if (idx < Np * Hv) {
        const int n = idx / Hv;
        const int k = idx % Hv;
        Wt[idx] = (n < Lv) ? W[(size_t)k * Lv + n] : 0.f;
    }
    if (blockIdx.x == 0 && threadIdx.x < Np)
        biasp[threadIdx.x] = (threadIdx.x < Lv) ? bias[threadIdx.x] : 0.f;
}

// -------------------------------------------------------------------------
// Kernel 1: emissions = hidden @ W + b  via V_WMMA_F32_16X16X4_F32.
// One wave per 16-row M tile; two 16x16 N tiles (N padded 21 -> 32).
// A layout (32-bit 16x4): lane = half*16 + m ; v[0]=K(2*half), v[1]=K(2*half+1)
// B layout (32-bit 4x16): lane = half*16 + n ; v[0]=K(2*half), v[1]=K(2*half+1)
// C/D layout: vgpr r, lane half h: row = r + 8*h, col = lane&15
// Inner loop: 3x global_load_b64 + 2x v_wmma + 1 prefetch, branch-free.
// -------------------------------------------------------------------------
__global__ __launch_bounds__(256) void k_emis(const float* __restrict__ hidden,
                                              const float* __restrict__ Wt,
                                              const float* __restrict__ biasp,
                                              float* __restrict__ emis)
{
    const int wave = (blockIdx.x * blockDim.x + threadIdx.x) >> 5; // M tile id
    const int lane = threadIdx.x & 31;
    const int half = lane >> 4;
    const int l16  = lane & 15;
    const int M0   = wave << 4;
    const int n1   = 16 + l16;

    // All three stream pointers include the +2*half K-pair offset (8B aligned).
    const float* arow = hidden + (size_t)(M0 + l16) * Hv + 2 * half;
    const float* b0p  = Wt + (size_t)l16 * Hv + 2 * half;
    const float* b1p  = Wt + (size_t)n1  * Hv + 2 * half;

    v8f acc0 = {};
    v8f acc1 = {};

    for (int k = 0; k < Hv; k += 4) {
        __builtin_prefetch(arow + k + 64, 0, 2); // ~256B ahead on the A stream
        const v2f a  = *(const v2f*)(arow + k);
        const v2f b0 = *(const v2f*)(b0p + k);
        const v2f b1 = *(const v2f*)(b1p + k);
        acc0 = __builtin_amdgcn_wmma_f32_16x16x4_f32(false, a, false, b0,
                                                     (short)0, acc0, false, false);
        acc1 = __builtin_amdgcn_wmma_f32_16x16x4_f32(false, a, false, b1,
                                                     (short)0, acc1, false, false);
    }

    const float bias0 = biasp[l16];
    const float bias1 = biasp[n1];
    const bool  n1ok  = (n1 < Lv);
    for (int r = 0; r < 8; ++r) {
        const int row = M0 + r + 8 * half;
        emis[(size_t)row * Lv + l16] = acc0[r] + bias0;
        if (n1ok) emis[(size_t)row * Lv + n1] = acc1[r] + bias1;
    }
}

// -------------------------------------------------------------------------
// Kernel 2: per-batch CRF logsumexp scan (denominator) + Viterbi forward
// and backtrack. One wave32 per batch; history kept in LDS (u8).
// Viterbi mask = attention_mask; CRF mask = attention_mask & (labels>=0).
// -------------------------------------------------------------------------
__global__ __launch_bounds__(32) void k_scan(const float* __restrict__ emis,
                                             const int* __restrict__ mask,
                                             const int* __restrict__ labels,
                                             const float* __restrict__ start,
                                             const float* __restrict__ endt,
                                             const float* __restrict__ trans,
                                             float* __restrict__ denom,
                                             int* __restrict__ tags)
{
    const int b = blockIdx.x;
    const int j = threadIdx.x;

    __shared__ float s_trans[Lv * Lv];
    __shared__ float s_score[Lv];
    __shared__ float s_v[Lv];
    __shared__ unsigned char s_hist[Sv * Lv]; // 10752 B

    for (int i = j; i < Lv * Lv; i += 32) s_trans[i] = trans[i];

    const float* eb = emis + (size_t)b * Sv * Lv;
    if (j < Lv) {
        const float s0 = start[j] + eb[j];
        s_score[j] = s0;
        s_v[j]     = s0;
    }
    __syncthreads();

    for (int t = 1; t < Sv; ++t) {
        float crf_new = 0.f, v_new = 0.f;
        if (j < Lv) {
            const float e = eb[(size_t)t * Lv + j];
            float m = -1e30f, best = -1e30f;
            int arg = 0;
            for (int i = 0; i < Lv; ++i) {
                const float tr = s_trans[i * Lv + j];
                m = fmaxf(m, s_score[i] + tr);
                const float cv = s_v[i] + tr;
                if (cv > best) { best = cv; arg = i; }
            }
            float ssum = 0.f;
            for (int i = 0; i < Lv; ++i)
                ssum += __expf(s_score[i] + s_trans[i * Lv + j] - m);
            crf_new = m + __logf(ssum) + e;
            v_new   = best + e;
            s_hist[t * Lv + j] = (unsigned char)arg;

            const int  mv    = mask[b * Sv + t];
            const bool m_vit = (mv != 0);
            const bool m_crf = m_vit && (labels[b * Sv + t] >= 0);
            if (!m_crf) crf_new = s_score[j];
            if (!m_vit) v_new   = s_v[j];
        }
        __syncthreads();
        if (j < Lv) { s_score[j] = crf_new; s_v[j] = v_new; }
        __syncthreads();
    }

    if (j == 0) {
        // denominator = logsumexp(score + end)
        float m = -1e30f;
        for (int i = 0; i < Lv; ++i) m = fmaxf(m, s_score[i] + endt[i]);
        float ssum = 0.f;
        for (int i = 0; i < Lv; ++i) ssum += __expf(s_score[i] + endt[i] - m);
        denom[b] = m + __logf(ssum);

        // Viterbi: best last tag, then backtrack through LDS history
        float best = -1e30f;
        int tag = 0;
        for (int i = 0; i < Lv; ++i) {
            const float c = s_v[i] + endt[i];
            if (c > best) { best = c; tag = i; }
        }
        for (int t = Sv - 1; t >= 1; --t) {
            tags[b * Sv + t] = tag;
            if (mask[b * Sv + t] != 0) tag = (int)s_hist[t * Lv + tag];
        }
        tags[b * Sv] = tag;
    }
}

// -------------------------------------------------------------------------
// Kernel 3: numerator + llh[b] = num - denom[b]
// tags from labels (clipped, tag[0]=0); valid = mask & labels>=0, valid[0]=1
// -------------------------------------------------------------------------
__global__ __launch_bounds__(256) void k_num(const float* __restrict__ emis,
                                             const int* __restrict__ mask,
                                             const int* __restrict__ labels,
                                             const float* __restrict__ start,
                                             const float* __restrict__ endt,
                                             const float* __restrict__ trans,
                                             const float* __restrict__ denom,
                                             float* __restrict__ llh)
{
    const int b   = blockIdx.x;
    const int tid = threadIdx.x;
    __shared__ float s_red[256];
    __shared__ int   s_cnt[256];

    float part = 0.f;
    int   cnt  = 0;
    for (int t = 1 + tid; t < Sv; t += 256) {
        const int lab = labels[b * Sv + t];
        const int tag = lab < 0 ? 0 : lab;
        int tagp;
        if (t - 1 == 0) {
            tagp = 0;
        } else {
            const int lp = labels[b * Sv + t - 1];
            tagp = lp < 0 ? 0 : lp;
        }
        const bool valid = (mask[b * Sv + t] != 0) && (lab >= 0);
        if (valid) {
            part += trans[tagp * Lv + tag] + emis[((size_t)b * Sv + t) * Lv + tag];
            cnt++;
        }
    }
    s_red[tid] = part;
    s_cnt[tid] = cnt;
    __syncthreads();
    for (int s = 128; s > 0; s >>= 1) {
        if (tid < s) { s_red[tid] += s_red[tid + s]; s_cnt[tid] += s_cnt[tid + s]; }
        __syncthreads();
    }
    if (tid == 0) {
        // tag0 = 0 (forced), valid[0] = true
        float num = start[0] + emis[(size_t)b * Sv * Lv + 0] + s_red[0];
        const int seq_end = s_cnt[0]; // (1 + cnt) - 1
        int last_tag;
        if (seq_end == 0) {
            last_tag = 0;
        } else {
            const int ll = labels[b * Sv + seq_end];
            last_tag = ll < 0 ? 0 : ll;
        }
        num += endt[last_tag];
        llh[b] = num - denom[b];
    }
}

// -------------------------------------------------------------------------
// Kernel 4: one-hot logits (masked) + loss = -mean(llh)
// -------------------------------------------------------------------------
__global__ __launch_bounds__(256) void k_out(const int* __restrict__ tags,
                                             const int* __restrict__ mask,
                                             const float* __restrict__ llh,
                                             float* __restrict__ out)
{
    const int idx = blockIdx.x * blockDim.x + threadIdx.x;
    if (idx < BS) {
        const int   tag = tags[idx];
        const float mf  = (mask[idx] != 0) ? 1.f : 0.f;
        float* o = out + 1 + (size_t)idx * Lv;
        for (int jj = 0; jj < Lv; ++jj) o[jj] = (jj == tag) ? mf : 0.f;
    }
    if (blockIdx.x == 0 && threadIdx.x == 0) {
        float s = 0.f;
        for (int i = 0; i < Bv; ++i) s += llh[i];
        out[0] = -s / (float)Bv;
    }
}

// -------------------------------------------------------------------------
extern "C" void kernel_launch(void* const* d_in, const int* in_sizes, int n_in,
                              void* d_out, int out_size, void* d_ws, size_t ws_size,
                              hipStream_t stream)
{
    (void)in_sizes; (void)n_in; (void)out_size; (void)ws_size;
    const float* hidden = (const float*)d_in[0];
    const int*   mask   = (const int*)  d_in[1];
    const int*   labels = (const int*)  d_in[2];
    const float* W      = (const float*)d_in[3];
    const float* bias   = (const float*)d_in[4];
    const float* start  = (const float*)d_in[5];
    const float* endt   = (const float*)d_in[6];
    const float* trans  = (const float*)d_in[7];
    float* out = (float*)d_out;

    char*  ws    = (char*)d_ws;
    float* emis  = (float*)ws;                              // BS*Lv floats
    float* denom = (float*)(ws + (size_t)BS * Lv * 4);      // Bv floats
    float* llh   = denom + Bv;                              // Bv floats
    int*   tags  = (int*)(llh + Bv);                        // BS ints
    float* Wt    = (float*)((char*)(tags + BS));            // Np*Hv floats
    float* biasp = Wt + (size_t)Np * Hv;                    // Np floats

    k_padw<<<(Np * Hv + 255) / 256, 256, 0, stream>>>(W, bias, Wt, biasp);
    // 2048 M-tiles, 8 waves (tiles) per 256-thread block -> 256 blocks
    k_emis<<<BS / 16 / 8, 256, 0, stream>>>(hidden, Wt, biasp, emis);
    k_scan<<<Bv, 32, 0, stream>>>(emis, mask, labels, start, endt, trans, denom, tags);
    k_num<<<Bv, 256, 0, stream>>>(emis, mask, labels, start, endt, trans, denom, llh);
    k_out<<<(BS + 255) / 256, 256, 0, stream>>>(tags, mask, llh, out);
}